// RGCN_25168508354749
// MI455X (gfx1250) — compile-verified
//
#include <hip/hip_runtime.h>
#include <hip/hip_bf16.h>

// ---------------------------------------------------------------------------
// RGCN 2-layer forward on gfx1250 (MI455X), fp32 throughout.
// Dense GEMM pieces use V_WMMA_F32_16X16X4_F32 (full fp32 precision).
// Aggregation uses exact (bit-monotonic) float atomic max -> deterministic.
// ---------------------------------------------------------------------------

typedef __attribute__((ext_vector_type(2))) float v2f;
typedef __attribute__((ext_vector_type(8))) float v8f;

#define N_NODES 100000
#define N_REL   7
#define IN_CH   128
#define HID_CH  64
#define OUT_CH  2
#define NEG_INF_BITS 0xFF800000u

// Exact float atomic-max via monotonic integer mapping (order independent).
__device__ __forceinline__ void atomicMaxFloat(float* addr, float val) {
    if (val >= 0.0f) {
        atomicMax((int*)addr, __float_as_int(val));
    } else {
        atomicMin((unsigned int*)addr, __float_as_uint(val));
    }
}

// ---------------------------------------------------------------------------
// Pass 0: zero relation counters
__global__ void k_zero_counts(int* counts) {
    if (threadIdx.x < N_REL + 1) counts[threadIdx.x] = 0;
}

// Pass 1: build per-relation compacted edge lists (lists[r*E + slot] = e)
__global__ void k_build_lists(const int* __restrict__ edge_type,
                              int* __restrict__ lists, int* __restrict__ counts,
                              int n_edges) {
    int e = blockIdx.x * blockDim.x + threadIdx.x;
    if (e >= n_edges) return;
    int r = edge_type[e];
    int slot = atomicAdd(&counts[r], 1);
    lists[(size_t)r * n_edges + slot] = e;
}

// ---------------------------------------------------------------------------
// Layer 1 self-transform: h = x @ root1 + bias1   (100000 x 128 @ 128 x 64)
// One wave computes one 16x16 f32 tile via 32 chained wmma_f32_16x16x4_f32.
// A layout (ISA 7.12.2, 16x4 f32): lane l holds rows m=l&15, K = (l>>4)*2 + {0,1}
// B layout (4x16 f32):             lane l holds cols n=l&15, K = (l>>4)*2 + {0,1}
// C/D (16x16 f32): c[i] of lane l -> row i + 8*(l>>4), col l&15
__global__ void k_l1_node_gemm(const float* __restrict__ x,
                               const float* __restrict__ root1,
                               const float* __restrict__ bias1,
                               float* __restrict__ h, int n_nodes) {
    const int lane = threadIdx.x & 31;
    const int wave = threadIdx.x >> 5;      // 4 waves -> 4 column tiles of 16
    const int row0 = blockIdx.x * 16;
    const int col0 = wave * 16;
    const int m    = lane & 15;             // A row / B col within tile
    const int kh   = (lane >> 4) * 2;       // K sub-offset: 0 or 2

    int arow = row0 + m;
    if (arow >= n_nodes) arow = n_nodes - 1;          // clamp (stores masked)
    const float* ap = x + (size_t)arow * IN_CH + kh;

    v8f c = {};
#pragma unroll
    for (int k0 = 0; k0 < IN_CH; k0 += 4) {
        v2f a, b;
        a.x = ap[k0];
        a.y = ap[k0 + 1];
        b.x = root1[(size_t)(k0 + kh)     * HID_CH + col0 + m];
        b.y = root1[(size_t)(k0 + kh + 1) * HID_CH + col0 + m];
        c = __builtin_amdgcn_wmma_f32_16x16x4_f32(false, a, false, b,
                                                  (short)0, c, false, false);
    }

    const int   half = (lane >> 4) * 8;
    const float bv   = bias1[col0 + m];
#pragma unroll
    for (int i = 0; i < 8; ++i) {
        int r = row0 + half + i;
        if (r < n_nodes)
            h[(size_t)r * HID_CH + col0 + m] = c[i] + bv;
    }
}

// ---------------------------------------------------------------------------
// Init a float buffer to -inf bit pattern (vectorized x4; n4 = n/4)
__global__ void k_init_neginf4(float4* __restrict__ p, size_t n4) {
    size_t i = (size_t)blockIdx.x * blockDim.x + threadIdx.x;
    if (i >= n4) return;
    float ni = __int_as_float((int)NEG_INF_BITS);
    p[i] = make_float4(ni, ni, ni, ni);
}

// ---------------------------------------------------------------------------
// Layer 1 edge messages for relation r, WMMA-batched 16 edges per wave.
// msg[16 edges, 64] = blockdiag: for b in 0..3: X[:, b*32:(b+1)*32] @ W1[r][b] (32x16)
// Each block-product = 8 chained wmma_f32_16x16x4_f32. Scatter via atomic max.
__global__ void k_l1_edge_wmma(const float* __restrict__ x,
                               const float* __restrict__ W1,
                               const int*   __restrict__ edge_index,
                               const int*   __restrict__ lists,
                               const int*   __restrict__ counts,
                               float* __restrict__ agg,
                               int rel, int n_edges) {
    const int lane  = threadIdx.x & 31;
    const int wave  = threadIdx.x >> 5;
    const int group = blockIdx.x * 4 + wave;       // 16 edges per group
    const int cnt   = counts[rel];
    if (group * 16 >= cnt) return;                 // wave-uniform exit

    const int m    = lane & 15;
    const int kh   = (lane >> 4) * 2;              // 0 or 2
    const int half = (lane >> 4) * 8;

    // Row m's edge (clamped so padded lanes load valid memory; masked at store)
    const int idx  = group * 16 + m;
    const int cidx = idx < cnt ? idx : cnt - 1;
    const int e    = lists[(size_t)rel * n_edges + cidx];
    const int src  = edge_index[e];
    const int dst_own = edge_index[n_edges + e];   // dst of row m (this lane)
    const float* xs = x + (size_t)src * IN_CH;

    // Scatter destinations for the 8 rows this lane owns (rows half..half+7):
    // fetched cross-lane (lanes 0..15 hold rows 0..15), block-invariant.
    int  dsts[8];
    bool dval[8];
#pragma unroll
    for (int i = 0; i < 8; ++i) {
        dsts[i] = __shfl(dst_own, half + i, 32);
        dval[i] = (group * 16 + half + i) < cnt;
    }

    const float* wr = W1 + (size_t)rel * 4 * 32 * 16;   // W1[rel]: [4][32][16]

#pragma unroll
    for (int b = 0; b < 4; ++b) {
        const float* wb = wr + (size_t)b * 32 * 16;     // [32][16]
        v8f c = {};
#pragma unroll
        for (int t = 0; t < 8; ++t) {
            int k = t * 4 + kh;                         // K within block (0..31)
            v2f a, bb;
            a.x  = xs[b * 32 + k];
            a.y  = xs[b * 32 + k + 1];
            bb.x = wb[(size_t)k       * 16 + m];
            bb.y = wb[(size_t)(k + 1) * 16 + m];
            c = __builtin_amdgcn_wmma_f32_16x16x4_f32(false, a, false, bb,
                                                      (short)0, c, false, false);
        }
        // Scatter rows half..half+7, column b*16 + m, into agg[dst]
#pragma unroll
        for (int i = 0; i < 8; ++i) {
            if (dval[i])
                atomicMaxFloat(&agg[(size_t)dsts[i] * HID_CH + b * 16 + m], c[i]);
        }
    }
}

// h += (agg finite ? agg : 0); optional ReLU on last relation. x4 vectorized.
__global__ void k_l1_accum4(float4* __restrict__ h, const float4* __restrict__ agg,
                            size_t n4, int do_relu) {
    size_t i = (size_t)blockIdx.x * blockDim.x + threadIdx.x;
    if (i >= n4) return;
    float4 a = agg[i];
    float4 v = h[i];
    if (__float_as_uint(a.x) != NEG_INF_BITS) v.x += a.x;
    if (__float_as_uint(a.y) != NEG_INF_BITS) v.y += a.y;
    if (__float_as_uint(a.z) != NEG_INF_BITS) v.z += a.z;
    if (__float_as_uint(a.w) != NEG_INF_BITS) v.w += a.w;
    if (do_relu) {
        v.x = fmaxf(v.x, 0.0f); v.y = fmaxf(v.y, 0.0f);
        v.z = fmaxf(v.z, 0.0f); v.w = fmaxf(v.w, 0.0f);
    }
    h[i] = v;
}

// ---------------------------------------------------------------------------
// W2[r] = sum_b comp2[r,b] * basis2[b]    -> [7][64][2]
__global__ void k_make_w2(const float* __restrict__ comp2,
                          const float* __restrict__ basis2,
                          float* __restrict__ W2) {
    int t = blockIdx.x * blockDim.x + threadIdx.x;
    if (t >= N_REL * HID_CH * OUT_CH) return;
    int r = t / (HID_CH * OUT_CH);
    int rem = t - r * (HID_CH * OUT_CH);
    int i = rem >> 1, o = rem & 1;
    float s = 0.0f;
#pragma unroll
    for (int b = 0; b < 4; ++b)
        s += comp2[r * 4 + b] * basis2[((size_t)b * HID_CH + i) * OUT_CH + o];
    W2[t] = s;
}

// Layer 2 edge pass: all relations in one scan, agg2 is [7][N][2].
// h rows and [64][2] weights read as float4 (b128 loads).
__global__ void k_l2_edge(const float* __restrict__ h,
                          const float* __restrict__ W2,
                          const int* __restrict__ edge_index,
                          const int* __restrict__ edge_type,
                          float* __restrict__ agg2,
                          int n_edges, int n_nodes) {
    int e = blockIdx.x * blockDim.x + threadIdx.x;
    if (e >= n_edges) return;
    int r = edge_type[e];
    int s = edge_index[e];
    int d = edge_index[n_edges + e];
    const float4* hr4 = (const float4*)(h + (size_t)s * HID_CH);
    const float4* w4  = (const float4*)(W2 + (size_t)r * HID_CH * OUT_CH);
    float m0 = 0.0f, m1 = 0.0f;
#pragma unroll
    for (int i = 0; i < HID_CH / 4; ++i) {
        float4 hv = hr4[i];
        float4 wa = w4[i * 2];       // w[4i+0],w[4i+1] interleaved (o0,o1)
        float4 wb = w4[i * 2 + 1];   // w[4i+2],w[4i+3]
        m0 += hv.x * wa.x + hv.y * wa.z + hv.z * wb.x + hv.w * wb.z;
        m1 += hv.x * wa.y + hv.y * wa.w + hv.z * wb.y + hv.w * wb.w;
    }
    float* slot = agg2 + ((size_t)r * n_nodes + d) * OUT_CH;
    atomicMaxFloat(slot,     m0);
    atomicMaxFloat(slot + 1, m1);
}

// Final: out = h @ root2 + bias2 + sum_r finite(agg2[r])
__global__ void k_l2_final(const float* __restrict__ h,
                           const float* __restrict__ root2,
                           const float* __restrict__ bias2,
                           const float* __restrict__ agg2,
                           float* __restrict__ out, int n_nodes) {
    int v = blockIdx.x * blockDim.x + threadIdx.x;
    if (v >= n_nodes) return;
    const float4* hr4 = (const float4*)(h + (size_t)v * HID_CH);
    const float4* w4  = (const float4*)root2;   // [64][2] interleaved
    float o0 = bias2[0], o1 = bias2[1];
#pragma unroll
    for (int i = 0; i < HID_CH / 4; ++i) {
        float4 hv = hr4[i];
        float4 wa = w4[i * 2];
        float4 wb = w4[i * 2 + 1];
        o0 += hv.x * wa.x + hv.y * wa.z + hv.z * wb.x + hv.w * wb.z;
        o1 += hv.x * wa.y + hv.y * wa.w + hv.z * wb.y + hv.w * wb.w;
    }
#pragma unroll
    for (int r = 0; r < N_REL; ++r) {
        const float* slot = agg2 + ((size_t)r * n_nodes + v) * OUT_CH;
        float a0 = slot[0], a1 = slot[1];
        if (__float_as_uint(a0) != NEG_INF_BITS) o0 += a0;
        if (__float_as_uint(a1) != NEG_INF_BITS) o1 += a1;
    }
    out[(size_t)v * OUT_CH]     = o0;
    out[(size_t)v * OUT_CH + 1] = o1;
}

// ---------------------------------------------------------------------------
extern "C" void kernel_launch(void* const* d_in, const int* in_sizes, int n_in,
                              void* d_out, int out_size, void* d_ws, size_t ws_size,
                              hipStream_t stream) {
    const float* x          = (const float*)d_in[0];
    const int*   edge_index = (const int*)  d_in[1];
    const int*   edge_type  = (const int*)  d_in[2];
    const float* W1         = (const float*)d_in[3];
    const float* root1      = (const float*)d_in[4];
    const float* bias1      = (const float*)d_in[5];
    const float* comp2      = (const float*)d_in[6];
    const float* basis2     = (const float*)d_in[7];
    const float* root2      = (const float*)d_in[8];
    const float* bias2      = (const float*)d_in[9];
    float*       out        = (float*)d_out;

    const int N = in_sizes[0] / IN_CH;   // 100000
    const int E = in_sizes[2];           // 1000000

    // Workspace carve-up (all fp32 / int32; segments stay 16B aligned)
    float* h     = (float*)d_ws;                         // N*64
    float* agg   = h   + (size_t)N * HID_CH;             // N*64
    float* agg2  = agg + (size_t)N * HID_CH;             // 7*N*2
    float* W2    = agg2 + (size_t)N_REL * N * OUT_CH;    // 7*64*2
    int*   counts = (int*)(W2 + N_REL * HID_CH * OUT_CH);// 8 (+pad to 32)
    int*   lists  = counts + 32;                         // 7*E

    // --- build per-relation edge lists -----------------------------------
    k_zero_counts<<<1, 32, 0, stream>>>(counts);
    k_build_lists<<<(E + 255) / 256, 256, 0, stream>>>(edge_type, lists, counts, E);

    // --- layer 1 self transform (WMMA fp32) ------------------------------
    k_l1_node_gemm<<<(N + 15) / 16, 128, 0, stream>>>(x, root1, bias1, h, N);

    // --- layer 1 per-relation messages + max aggregation -----------------
    const size_t nh4 = (size_t)N * HID_CH / 4;
    const int groups_max  = (E + 15) / 16;               // worst case all-one-rel
    const int edge_blocks = (groups_max + 3) / 4;        // 4 waves per block
    for (int r = 0; r < N_REL; ++r) {
        k_init_neginf4<<<(unsigned)((nh4 + 255) / 256), 256, 0, stream>>>(
            (float4*)agg, nh4);
        k_l1_edge_wmma<<<edge_blocks, 128, 0, stream>>>(x, W1, edge_index,
                                                        lists, counts, agg, r, E);
        k_l1_accum4<<<(unsigned)((nh4 + 255) / 256), 256, 0, stream>>>(
            (float4*)h, (const float4*)agg, nh4, r == N_REL - 1 ? 1 : 0);
    }

    // --- layer 2 ---------------------------------------------------------
    k_make_w2<<<(N_REL * HID_CH * OUT_CH + 127) / 128, 128, 0, stream>>>(comp2, basis2, W2);
    const size_t na24 = (size_t)N_REL * N * OUT_CH / 4;
    k_init_neginf4<<<(unsigned)((na24 + 255) / 256), 256, 0, stream>>>(
        (float4*)agg2, na24);
    k_l2_edge<<<(E + 255) / 256, 256, 0, stream>>>(h, W2, edge_index, edge_type,
                                                   agg2, E, N);
    k_l2_final<<<(N + 255) / 256, 256, 0, stream>>>(h, root2, bias2, agg2, out, N);
}